// CloverWilsonDirac_54623394071202
// MI455X (gfx1250) — compile-verified
//
#include <hip/hip_runtime.h>

// ---------------------------------------------------------------------------
// Clover-Wilson Dirac operator on a 32x24x24x24 lattice, NC=3, NS=4, fp32
// complex.  One thread = one site.  Field strength / Wilson hop in VALU fp32
// FMA chains; the sigma_{mu,nu} (x) F contraction is a real-embedded GEMM
// D[8x48]x[48xN] executed on the matrix cores with chained
// v_wmma_f32_16x16x4_f32 (wave32, K=48 -> 12 WMMAs per 16-site tile, x3
// colors, x2 site groups = 72 WMMAs per wave).
// ---------------------------------------------------------------------------

namespace {

constexpr int LT = 32, LZ = 24, LY = 24, LX = 24;
constexpr int VOL = LT * LZ * LY * LX;     // 442368
constexpr float MASSV = 0.1f;
constexpr float CSWV  = 1.0f;

typedef __attribute__((ext_vector_type(2))) float v2f;
typedef __attribute__((ext_vector_type(8))) float v8f;

struct cx { float re, im; };

__device__ __forceinline__ cx cadd(cx a, cx b) { return {a.re + b.re, a.im + b.im}; }
__device__ __forceinline__ cx cscale(cx a, float s) { return {a.re * s, a.im * s}; }
// acc += a*b
__device__ __forceinline__ cx cmac(cx acc, cx a, cx b) {
  acc.re = fmaf(a.re, b.re, fmaf(-a.im, b.im, acc.re));
  acc.im = fmaf(a.re, b.im, fmaf( a.im, b.re, acc.im));
  return acc;
}
// acc += a*conj(b)
__device__ __forceinline__ cx cmacj(cx acc, cx a, cx b) {
  acc.re = fmaf(a.re, b.re, fmaf( a.im, b.im, acc.re));
  acc.im = fmaf(a.im, b.re, fmaf(-a.re, b.im, acc.im));
  return acc;
}
// acc += conj(a)*b
__device__ __forceinline__ cx cjmac(cx acc, cx a, cx b) {
  acc.re = fmaf(a.re, b.re, fmaf( a.im, b.im, acc.re));
  acc.im = fmaf(a.re, b.im, fmaf(-a.im, b.re, acc.im));
  return acc;
}
__device__ __forceinline__ cx imul(cx v)  { return {-v.im,  v.re}; }  //  i*v
__device__ __forceinline__ cx mimul(cx v) { return { v.im, -v.re}; }  // -i*v
__device__ __forceinline__ cx cneg(cx v)  { return {-v.re, -v.im}; }

struct M3 { cx e[3][3]; };

__device__ __forceinline__ M3 loadU(const cx* __restrict__ U, int site, int mu) {
  M3 m;
  const cx* b = U + (site * 4 + mu) * 9;
#pragma unroll
  for (int i = 0; i < 3; ++i)
#pragma unroll
    for (int j = 0; j < 3; ++j) m.e[i][j] = b[i * 3 + j];
  return m;
}

__device__ __forceinline__ M3 mmul(const M3& A, const M3& B) {  // A*B
  M3 R;
#pragma unroll
  for (int i = 0; i < 3; ++i)
#pragma unroll
    for (int j = 0; j < 3; ++j) {
      cx a = {0.f, 0.f};
      a = cmac(a, A.e[i][0], B.e[0][j]);
      a = cmac(a, A.e[i][1], B.e[1][j]);
      a = cmac(a, A.e[i][2], B.e[2][j]);
      R.e[i][j] = a;
    }
  return R;
}

__device__ __forceinline__ M3 mmul_dagB(const M3& A, const M3& B) {  // A*B^dag
  M3 R;
#pragma unroll
  for (int i = 0; i < 3; ++i)
#pragma unroll
    for (int j = 0; j < 3; ++j) {
      cx a = {0.f, 0.f};
      a = cmacj(a, A.e[i][0], B.e[j][0]);
      a = cmacj(a, A.e[i][1], B.e[j][1]);
      a = cmacj(a, A.e[i][2], B.e[j][2]);
      R.e[i][j] = a;
    }
  return R;
}

// Q += P  and  Q += P^dag
__device__ __forceinline__ void madd(M3& Q, const M3& P) {
#pragma unroll
  for (int i = 0; i < 3; ++i)
#pragma unroll
    for (int j = 0; j < 3; ++j) Q.e[i][j] = cadd(Q.e[i][j], P.e[i][j]);
}
__device__ __forceinline__ void madd_dag(M3& Q, const M3& P) {
#pragma unroll
  for (int i = 0; i < 3; ++i)
#pragma unroll
    for (int j = 0; j < 3; ++j) {
      Q.e[i][j].re += P.e[j][i].re;
      Q.e[i][j].im -= P.e[j][i].im;
    }
}

// (G1*G2)*G3^dag*G4^dag  (one clover leaf)
__device__ __forceinline__ M3 staple(const M3& A, const M3& B, const M3& C, const M3& D) {
  M3 P = mmul(A, B);
  P = mmul_dagB(P, C);
  P = mmul_dagB(P, D);
  return P;
}

__device__ __forceinline__ int wrapc(int v, int n) {
  if (v < 0) v += n; else if (v >= n) v -= n;
  return v;
}

// site index of x + dmu*mu_hat + dnu*nu_hat (dims order: t,z,y,x)
__device__ __forceinline__ int shifted_idx(int t, int z, int y, int x,
                                           int mu, int dmu, int nu, int dnu) {
  int c[4] = {t, z, y, x};
  const int D[4] = {LT, LZ, LY, LX};
  c[mu] = wrapc(c[mu] + dmu, D[mu]);
  c[nu] = wrapc(c[nu] + dnu, D[nu]);
  return ((c[0] * LZ + c[1]) * LY + c[2]) * LX + c[3];
}

// gamma_mu * h (spin mixing), DeGrand-Rossi basis of the reference
__device__ __forceinline__ void gamma_mul(int mu, const cx h[4][3], cx g[4][3]) {
#pragma unroll
  for (int c = 0; c < 3; ++c) {
    switch (mu) {
      case 0: g[0][c] = h[2][c];        g[1][c] = h[3][c];
              g[2][c] = h[0][c];        g[3][c] = h[1][c];        break;
      case 1: g[0][c] = imul(h[3][c]);  g[1][c] = imul(h[2][c]);
              g[2][c] = mimul(h[1][c]); g[3][c] = mimul(h[0][c]); break;
      case 2: g[0][c] = cneg(h[3][c]);  g[1][c] = h[2][c];
              g[2][c] = h[1][c];        g[3][c] = cneg(h[0][c]);  break;
      default:g[0][c] = imul(h[2][c]);  g[1][c] = mimul(h[3][c]);
              g[2][c] = mimul(h[0][c]); g[3][c] = imul(h[1][c]);  break;
    }
  }
}

// sigma_{mu,nu} = i*g_mu*g_nu (block-diagonal in spin), split re/im, entries 0,+-1
__device__ __constant__ float SRE[6][4][4] = {
  {{0,1,0,0},{1,0,0,0},{0,0,0,-1},{0,0,-1,0}},   // (0,1)
  {{0,0,0,0},{0,0,0,0},{0,0,0,0},{0,0,0,0}},     // (0,2) pure imag
  {{1,0,0,0},{0,-1,0,0},{0,0,-1,0},{0,0,0,1}},   // (0,3)
  {{1,0,0,0},{0,-1,0,0},{0,0,1,0},{0,0,0,-1}},   // (1,2)
  {{0,0,0,0},{0,0,0,0},{0,0,0,0},{0,0,0,0}},     // (1,3) pure imag
  {{0,1,0,0},{1,0,0,0},{0,0,0,1},{0,0,1,0}},     // (2,3)
};
__device__ __constant__ float SIM[6][4][4] = {
  {{0,0,0,0},{0,0,0,0},{0,0,0,0},{0,0,0,0}},
  {{0,1,0,0},{-1,0,0,0},{0,0,0,-1},{0,0,1,0}},
  {{0,0,0,0},{0,0,0,0},{0,0,0,0},{0,0,0,0}},
  {{0,0,0,0},{0,0,0,0},{0,0,0,0},{0,0,0,0}},
  {{0,-1,0,0},{1,0,0,0},{0,0,0,-1},{0,0,1,0}},
  {{0,0,0,0},{0,0,0,0},{0,0,0,0},{0,0,0,0}},
};

// Real embedding S'[m][k]: m=(spin a, re/im), k=(plane p, spin b, re/im)
__device__ __forceinline__ float sprime(int m, int k) {
  const int a = m >> 1, rr = m & 1;
  const int p = k >> 3, b = (k >> 1) & 3, r = k & 1;
  const float sre = SRE[p][a][b], sim = SIM[p][a][b];
  return (rr == 0) ? ((r == 0) ? sre : -sim)
                   : ((r == 0) ? sim :  sre);
}

__global__ __launch_bounds__(64)
void clover_wilson_kernel(const cx* __restrict__ psi,
                          const cx* __restrict__ Ug,
                          cx* __restrict__ outg) {
  // chi_p[b][color] per site: X operand of the sigma-GEMM
  __shared__ cx    chiS[64][6][4][3];     // 36 KB
  __shared__ float outS[64][4][3][2];     //  6 KB

  const int ls = threadIdx.x;
  const int s  = blockIdx.x * 64 + ls;
  const int x  = s % LX;
  const int y  = (s / LX) % LY;
  const int z  = (s / (LX * LY)) % LZ;
  const int t  = s / (LX * LY * LZ);

  __builtin_prefetch(&Ug[s * 36], 0, 3);
  __builtin_prefetch(&psi[s * 12], 0, 3);

  // ---- load psi(x) ----
  cx ps[4][3];
#pragma unroll
  for (int a = 0; a < 4; ++a)
#pragma unroll
    for (int c = 0; c < 3; ++c) ps[a][c] = psi[s * 12 + a * 3 + c];

  // ---- diagonal: (4+m)*psi from Wilson + 1*psi from clover identity ----
  cx acc[4][3];
#pragma unroll
  for (int a = 0; a < 4; ++a)
#pragma unroll
    for (int c = 0; c < 3; ++c) acc[a][c] = cscale(ps[a][c], 5.0f + MASSV);

  // ---- Wilson hop ----
  for (int mu = 0; mu < 4; ++mu) {
    const int sf = shifted_idx(t, z, y, x, mu,  1, mu, 0);
    const int sb = shifted_idx(t, z, y, x, mu, -1, mu, 0);
    const M3 Uf = loadU(Ug, s,  mu);
    const M3 Ub = loadU(Ug, sb, mu);
    cx hf[4][3], hb[4][3];
#pragma unroll
    for (int b = 0; b < 4; ++b) {
      cx vf[3], vb[3];
#pragma unroll
      for (int j = 0; j < 3; ++j) {
        vf[j] = psi[sf * 12 + b * 3 + j];
        vb[j] = psi[sb * 12 + b * 3 + j];
      }
#pragma unroll
      for (int i = 0; i < 3; ++i) {
        cx af = {0.f, 0.f}, ab = {0.f, 0.f};
#pragma unroll
        for (int j = 0; j < 3; ++j) {
          af = cmac (af, Uf.e[i][j], vf[j]);   // U(x) psi(x+mu)
          ab = cjmac(ab, Ub.e[j][i], vb[j]);   // U(x-mu)^dag psi(x-mu)
        }
        hf[b][i] = af; hb[b][i] = ab;
      }
    }
    cx gf[4][3], gb[4][3];
    gamma_mul(mu, hf, gf);
    gamma_mul(mu, hb, gb);
#pragma unroll
    for (int a = 0; a < 4; ++a)
#pragma unroll
      for (int c = 0; c < 3; ++c) {
        acc[a][c].re -= 0.5f * (hf[a][c].re - gf[a][c].re + hb[a][c].re + gb[a][c].re);
        acc[a][c].im -= 0.5f * (hf[a][c].im - gf[a][c].im + hb[a][c].im + gb[a][c].im);
      }
  }

  // ---- field strength per plane, chi_p[b] = F_p * psi_b -> LDS ----
  const int PMU[6] = {0, 0, 0, 1, 1, 2};
  const int PNU[6] = {1, 2, 3, 2, 3, 3};
  for (int p = 0; p < 6; ++p) {
    const int mu = PMU[p], nu = PNU[p];
    M3 Q;
    {  // leaf 1: U_mu(x) U_nu(x+mu) U_mu(x+nu)^ U_nu(x)^
      M3 A = loadU(Ug, s, mu);
      M3 B = loadU(Ug, shifted_idx(t, z, y, x, mu, 1, nu, 0), nu);
      M3 C = loadU(Ug, shifted_idx(t, z, y, x, mu, 0, nu, 1), mu);
      M3 D = loadU(Ug, s, nu);
      Q = staple(A, B, C, D);
    }
    {  // leaf 2 (reference roll composition)
      M3 A = loadU(Ug, shifted_idx(t, z, y, x, mu, -1, nu, 1), nu);
      M3 B = loadU(Ug, shifted_idx(t, z, y, x, mu, -1, nu, 2), mu);
      M3 C = loadU(Ug, shifted_idx(t, z, y, x, mu,  0, nu, 1), nu);
      M3 D = loadU(Ug, shifted_idx(t, z, y, x, mu, -1, nu, 1), mu);
      madd(Q, staple(A, B, C, D));
    }
    {  // leaf 3 (daggered)
      M3 A = loadU(Ug, shifted_idx(t, z, y, x, mu, -2, nu, -2), mu);
      M3 B = loadU(Ug, shifted_idx(t, z, y, x, mu, -1, nu, -2), nu);
      M3 C = loadU(Ug, shifted_idx(t, z, y, x, mu, -2, nu, -1), mu);
      M3 D = loadU(Ug, shifted_idx(t, z, y, x, mu, -2, nu, -2), nu);
      madd_dag(Q, staple(A, B, C, D));
    }
    {  // leaf 4 (daggered)
      M3 A = loadU(Ug, shifted_idx(t, z, y, x, mu, 0, nu, -2), mu);
      M3 B = loadU(Ug, shifted_idx(t, z, y, x, mu, 1, nu, -2), nu);
      M3 C = loadU(Ug, shifted_idx(t, z, y, x, mu, 0, nu, -1), mu);
      M3 D = loadU(Ug, shifted_idx(t, z, y, x, mu, 0, nu, -2), nu);
      madd_dag(Q, staple(A, B, C, D));
    }
    // F = (Q - Q^dag)/(8i):  F_ij = ((Qij.im+Qji.im)/8, -(Qij.re-Qji.re)/8)
    M3 F;
#pragma unroll
    for (int i = 0; i < 3; ++i)
#pragma unroll
      for (int j = 0; j < 3; ++j) {
        F.e[i][j].re =  (Q.e[i][j].im + Q.e[j][i].im) * 0.125f;
        F.e[i][j].im = -(Q.e[i][j].re - Q.e[j][i].re) * 0.125f;
      }
#pragma unroll
    for (int b = 0; b < 4; ++b)
#pragma unroll
      for (int i = 0; i < 3; ++i) {
        cx a = {0.f, 0.f};
        a = cmac(a, F.e[i][0], ps[b][0]);
        a = cmac(a, F.e[i][1], ps[b][1]);
        a = cmac(a, F.e[i][2], ps[b][2]);
        chiS[ls][p][b][i] = a;
      }
  }
  __syncthreads();

  // ---- sigma contraction as real GEMM on matrix cores --------------------
  // D[8 x 16sites] = S'[8 x 48] * X[48 x 16sites], chained K=4 f32 WMMA.
  // A 16x4 layout: lanes0-15 rows, VGPR{0,1}=K{0,1}; lanes16-31: K{2,3}.
  // B 4x16 layout: lanes0-15 cols w/ rows{0,1}; lanes16-31 same cols rows{2,3}.
  const int lane  = ls & 31;
  const int wbase = (ls >> 5) * 32;
  const int hi    = lane >> 4;
  const int col   = lane & 15;

  v2f afr[12];
#pragma unroll
  for (int kc = 0; kc < 12; ++kc) {
    const int k0 = kc * 4 + hi * 2;
    afr[kc][0] = (col < 8) ? sprime(col, k0)     : 0.f;
    afr[kc][1] = (col < 8) ? sprime(col, k0 + 1) : 0.f;
  }

  for (int g = 0; g < 2; ++g) {
    const int bls = wbase + g * 16 + col;   // site (column) this lane feeds
    for (int c = 0; c < 3; ++c) {
      v8f dacc = {0.f, 0.f, 0.f, 0.f, 0.f, 0.f, 0.f, 0.f};
#pragma unroll
      for (int kc = 0; kc < 12; ++kc) {
        const int k0 = kc * 4 + hi * 2;
        v2f bfr;
        {
          const int p = k0 >> 3, b = (k0 >> 1) & 3, r = k0 & 1;
          const cx v = chiS[bls][p][b][c];
          bfr[0] = r ? v.im : v.re;
        }
        {
          const int k1 = k0 + 1;
          const int p = k1 >> 3, b = (k1 >> 1) & 3, r = k1 & 1;
          const cx v = chiS[bls][p][b][c];
          bfr[1] = r ? v.im : v.re;
        }
        dacc = __builtin_amdgcn_wmma_f32_16x16x4_f32(
            false, afr[kc], false, bfr, (short)0, dacc, false, false);
      }
      if (lane < 16) {  // rows 0..7 (lanes 0-15, VGPRs 0-7) hold the result
        outS[bls][0][c][0] = dacc[0]; outS[bls][0][c][1] = dacc[1];
        outS[bls][1][c][0] = dacc[2]; outS[bls][1][c][1] = dacc[3];
        outS[bls][2][c][0] = dacc[4]; outS[bls][2][c][1] = dacc[5];
        outS[bls][3][c][0] = dacc[6]; outS[bls][3][c][1] = dacc[7];
      }
    }
  }
  __syncthreads();

  // ---- add (csw/4) * sigma-contraction and store -------------------------
#pragma unroll
  for (int a = 0; a < 4; ++a)
#pragma unroll
    for (int c = 0; c < 3; ++c) {
      cx r = acc[a][c];
      r.re += (CSWV * 0.25f) * outS[ls][a][c][0];
      r.im += (CSWV * 0.25f) * outS[ls][a][c][1];
      outg[s * 12 + a * 3 + c] = r;
    }
}

}  // namespace

extern "C" void kernel_launch(void* const* d_in, const int* in_sizes, int n_in,
                              void* d_out, int out_size, void* d_ws, size_t ws_size,
                              hipStream_t stream) {
  (void)in_sizes; (void)n_in; (void)out_size; (void)d_ws; (void)ws_size;
  const cx* psi = reinterpret_cast<const cx*>(d_in[0]);
  const cx* U   = reinterpret_cast<const cx*>(d_in[1]);
  cx* out       = reinterpret_cast<cx*>(d_out);
  clover_wilson_kernel<<<VOL / 64, 64, 0, stream>>>(psi, U, out);
}